// LargeLBCS_386547056739
// MI455X (gfx1250) — compile-verified
//
#include <hip/hip_runtime.h>
#include <hip/hip_bf16.h>
#include <math.h>

// Problem constants (match reference)
#define PN 100000
#define PS 64
#define PQ 50
#define PP 3
#define P_EPS 1e-12f

#define WORDS_PER_BLOCK 128                       // 8 waves * 16 words
#define NBLOCKS ((PN + WORDS_PER_BLOCK - 1) / WORDS_PER_BLOCK)  // 782

typedef __attribute__((ext_vector_type(2))) float v2f;
typedef __attribute__((ext_vector_type(8))) float v8f;

__device__ __forceinline__ float softplus20(float x) {
    float z = 20.0f * x;
    // numerically stable softplus
    return (z > 0.0f) ? (z + log1pf(expf(-z))) : log1pf(expf(z));
}

// ---------------------------------------------------------------------------
// Kernel 1: tiny prep — normalized heads (padded to 4 floats) + head ratios
// ---------------------------------------------------------------------------
__global__ __launch_bounds__(256) void lbcs_prep_kernel(
    const float* __restrict__ heads_param,   // [S,Q,3]
    const float* __restrict__ hr_param,      // [S]
    float* __restrict__ ws_heads,            // [S,Q,4] padded
    float* __restrict__ ws_hr)               // [S]
{
    const int t = threadIdx.x;

    // heads: softplus(20x), L1-normalize over p, pad 4th component with 0
    for (int i = t; i < PS * PQ; i += 256) {
        const float* hp = heads_param + (size_t)i * PP;
        float a0 = softplus20(hp[0]);
        float a1 = softplus20(hp[1]);
        float a2 = softplus20(hp[2]);
        float inv = 1.0f / fmaxf(a0 + a1 + a2, P_EPS);
        float* o = ws_heads + (size_t)i * 4;
        o[0] = a0 * inv;
        o[1] = a1 * inv;
        o[2] = a2 * inv;
        o[3] = 0.0f;                 // K=3 pad for the 16x16x4 WMMA
    }

    // head ratios
    __shared__ float sh[PS];
    __shared__ float ssum;
    float v = 0.0f;
    if (t < PS) {
        v = softplus20(hr_param[t]);
        sh[t] = v;
    }
    __syncthreads();
    if (t == 0) {
        float s = 0.0f;
        for (int i = 0; i < PS; ++i) s += sh[i];
        ssum = fmaxf(s, P_EPS);
    }
    __syncthreads();
    if (t < PS) {
        float r = v / ssum;
        ws_hr[t] = (r + 0.001f / (float)PS) / 1.001f;
    }
}

// ---------------------------------------------------------------------------
// Kernel 2: main — per q WMMA dot (16 words x 16 heads), product over q,
// ratio-weighted row sums, coeff^2/cov partial per block.
// ---------------------------------------------------------------------------
__global__ __launch_bounds__(256) void lbcs_cov_kernel(
    const float* __restrict__ pauli,     // [N,Q,3]
    const float* __restrict__ coeff,     // [N]
    const float* __restrict__ ws_heads,  // [S,Q,4]
    const float* __restrict__ ws_hr,     // [S]
    float* __restrict__ partials)        // [NBLOCKS]
{
    __shared__ float lh[PS * PQ * 4];    // 12800 floats = 50 KB
    __shared__ float lhr[PS];
    __shared__ float lred[16];

    const int t = threadIdx.x;
    for (int i = t; i < PS * PQ * 4; i += 256) lh[i] = ws_heads[i];
    if (t < PS) lhr[t] = ws_hr[t];
    __syncthreads();

    const int wave = t >> 5;
    const int lane = t & 31;
    const int hh   = lane >> 4;          // half-wave: 0 -> K=0,1 ; 1 -> K=2,3
    const int lm   = lane & 15;
    const int k    = hh ? 2 : 0;

    const int n0    = blockIdx.x * WORDS_PER_BLOCK + wave * 16;
    const int rowN  = n0 + lm;
    const int nclmp = (rowN < PN) ? rowN : (PN - 1);
    const float* pbase = pauli + (size_t)nclmp * (PQ * PP);

    v8f prod[4];
    #pragma unroll
    for (int st = 0; st < 4; ++st)
        #pragma unroll
        for (int r = 0; r < 8; ++r) prod[st][r] = 1.0f;

    const v8f czero = {};

    #pragma unroll 2
    for (int q = 0; q < PQ; ++q) {
        // branch-free stream-ahead prefetch (address clamped into this row)
        int qp = q + 8;
        qp = (qp < PQ) ? qp : (PQ - 1);
        __builtin_prefetch(&pbase[qp * PP], 0, 1);

        // A tile: 16x4 f32 (lanes 0-15: K=0,1 ; lanes 16-31: K=2,3=0)
        v2f a;
        a.x = pbase[q * PP + k];
        float ay = pbase[q * PP + 1];
        a.y = hh ? 0.0f : ay;

        // 4 head tiles: B is 4x16, lane lm = head column, pair of K comps
        #pragma unroll
        for (int st = 0; st < 4; ++st) {
            const v2f b = *(const v2f*)&lh[(((st * 16 + lm) * PQ) + q) * 4 + k];
            v8f d = __builtin_amdgcn_wmma_f32_16x16x4_f32(
                false, a, false, b, (short)0, czero, false, false);
            #pragma unroll
            for (int r = 0; r < 8; ++r) prod[st][r] *= d[r];
        }
    }

    // ratio weighting + cross-lane row reduction (C layout: VGPR r, lane half)
    float res[8];
    #pragma unroll
    for (int r = 0; r < 8; ++r) {
        float val = prod[0][r] * lhr[lm]
                  + prod[1][r] * lhr[16 + lm]
                  + prod[2][r] * lhr[32 + lm]
                  + prod[3][r] * lhr[48 + lm];
        val += __shfl_xor(val, 1, 32);
        val += __shfl_xor(val, 2, 32);
        val += __shfl_xor(val, 4, 32);
        val += __shfl_xor(val, 8, 32);
        res[r] = val;                 // per-word cov, same in all 16 lanes of a half
    }

    float part = 0.0f;
    if (lm == 0) {
        #pragma unroll
        for (int r = 0; r < 8; ++r) {
            int n = n0 + r + hh * 8;  // VGPR r: half 0 -> M=r, half 1 -> M=r+8
            if (n < PN) {
                float c = coeff[n];
                part += c * c / res[r];
            }
        }
        lred[wave * 2 + hh] = part;
    }
    __syncthreads();
    if (t == 0) {
        float s = 0.0f;
        for (int i = 0; i < 16; ++i) s += lred[i];
        partials[blockIdx.x] = s;
    }
}

// ---------------------------------------------------------------------------
// Kernel 3: deterministic final reduction of per-block partials
// ---------------------------------------------------------------------------
__global__ __launch_bounds__(256) void lbcs_reduce_kernel(
    const float* __restrict__ partials, int nblocks, float* __restrict__ out)
{
    __shared__ float sh[256];
    const int t = threadIdx.x;
    float s = 0.0f;
    for (int i = t; i < nblocks; i += 256) s += partials[i];
    sh[t] = s;
    __syncthreads();
    for (int off = 128; off > 0; off >>= 1) {
        if (t < off) sh[t] += sh[t + off];
        __syncthreads();
    }
    if (t == 0) out[0] = sh[0];
}

// ---------------------------------------------------------------------------
// Launch
// ---------------------------------------------------------------------------
extern "C" void kernel_launch(void* const* d_in, const int* in_sizes, int n_in,
                              void* d_out, int out_size, void* d_ws, size_t ws_size,
                              hipStream_t stream) {
    const float* pauli   = (const float*)d_in[0];  // [N,Q,3]
    const float* coeff   = (const float*)d_in[1];  // [N]
    const float* heads_p = (const float*)d_in[2];  // [S,Q,3]
    const float* hr_p    = (const float*)d_in[3];  // [S]
    float* out = (float*)d_out;

    // workspace layout (floats)
    float* ws_heads = (float*)d_ws;                       // S*Q*4 = 12800
    float* ws_hr    = ws_heads + PS * PQ * 4;             // 64
    float* ws_part  = ws_hr + PS;                         // NBLOCKS

    lbcs_prep_kernel<<<1, 256, 0, stream>>>(heads_p, hr_p, ws_heads, ws_hr);
    lbcs_cov_kernel<<<NBLOCKS, 256, 0, stream>>>(pauli, coeff, ws_heads, ws_hr, ws_part);
    lbcs_reduce_kernel<<<1, 256, 0, stream>>>(ws_part, NBLOCKS, out);
}